// Attention_60833916781258
// MI455X (gfx1250) — compile-verified
//
#include <hip/hip_runtime.h>
#include <hip/hip_bf16.h>
#include <math.h>
#include <stdint.h>

// ---------------------------------------------------------------------------
// Problem constants (from reference setup_inputs)
// ---------------------------------------------------------------------------
#define B_   2
#define N_   4096
#define F_   128
#define U_   64
#define BN_  (B_ * N_)
#define NEG_BIG (-1e10f)
#define SLOPE   0.2f

// A-chunking for the async-LDS pipeline in the attention kernel
#define JCHUNK   64                    // j columns per chunk
#define NCHUNK   (N_ / JCHUNK)         // 64 chunks
#define SEGS     ((16 * JCHUNK * 4) / 16 / 32)   // b128 segs per lane per chunk = 8

typedef __attribute__((ext_vector_type(2))) float v2f;
typedef __attribute__((ext_vector_type(4))) float v4f;
typedef __attribute__((ext_vector_type(8))) float v8f;

__device__ __forceinline__ float lrelu(float x) {
    // leaky_relu, slope 0.2
    return fmaxf(x, 0.0f) + SLOPE * fminf(x, 0.0f);
}

// ---------------------------------------------------------------------------
// Kernel 1: X = H @ W   (per component re/im via blockIdx.y)
// One wave computes a 16x64 tile with V_WMMA_F32_16X16X4_F32.
// A-frag (16x4 f32): lanes 0-15 hold K=0,1 ; lanes 16-31 hold K=2,3 (M = lane&15)
// B-frag (4x16 f32): lanes 0-15 hold K=0,1 ; lanes 16-31 hold K=2,3 (Ncol = lane&15)
// C (16x16 f32, v8f): VGPR r -> row r (lanes 0-15) / row r+8 (lanes 16-31)
//
// Output is stored PAIR-INTERLEAVED for the attention pass:
//   Xp[(row>>1)*(2*U) + col*2 + (row&1)]
// so {X[j][u], X[j+1][u]} is one aligned 8-byte word (a B-fragment lane).
// ---------------------------------------------------------------------------
__global__ void __launch_bounds__(32)
xform_kernel(const float* __restrict__ Hre, const float* __restrict__ Him,
             const float* __restrict__ W,
             float* __restrict__ Xp_re, float* __restrict__ Xp_im) {
    const int lane = threadIdx.x & 31;
    const float* __restrict__ H = blockIdx.y ? Him : Hre;
    float* __restrict__ Xp      = blockIdx.y ? Xp_im : Xp_re;

    const int  i0   = blockIdx.x * 16;          // 16 rows of [B*N]
    const int  m    = lane & 15;
    const int  koff = (lane >> 4) * 2;
    const long rowB = (long)(i0 + m) * F_;

    v8f acc0 = {}, acc1 = {}, acc2 = {}, acc3 = {};

    for (int k = 0; k < F_; k += 4) {
        v2f a = *(const v2f*)(H + rowB + k + koff);           // H[i, k+koff .. +1]
        const float* w0 = W + (long)(k + koff) * U_ + m;      // W[k+koff  , :]
        const float* w1 = W + (long)(k + koff + 1) * U_ + m;  // W[k+koff+1, :]
        v2f b0 = { w0[0],  w1[0]  };
        v2f b1 = { w0[16], w1[16] };
        v2f b2 = { w0[32], w1[32] };
        v2f b3 = { w0[48], w1[48] };
        acc0 = __builtin_amdgcn_wmma_f32_16x16x4_f32(false, a, false, b0, (short)0, acc0, false, false);
        acc1 = __builtin_amdgcn_wmma_f32_16x16x4_f32(false, a, false, b1, (short)0, acc1, false, false);
        acc2 = __builtin_amdgcn_wmma_f32_16x16x4_f32(false, a, false, b2, (short)0, acc2, false, false);
        acc3 = __builtin_amdgcn_wmma_f32_16x16x4_f32(false, a, false, b3, (short)0, acc3, false, false);
    }

    const int rhi = (lane >> 4) * 8;
#pragma unroll
    for (int r = 0; r < 8; r += 2) {
        const int row = i0 + rhi + r;                         // even row
        float* base = Xp + (long)(row >> 1) * (2 * U_);
        v2f d0 = { acc0[r], acc0[r + 1] };
        v2f d1 = { acc1[r], acc1[r + 1] };
        v2f d2 = { acc2[r], acc2[r + 1] };
        v2f d3 = { acc3[r], acc3[r + 1] };
        *(v2f*)(base + ( 0 + m) * 2) = d0;
        *(v2f*)(base + (16 + m) * 2) = d1;
        *(v2f*)(base + (32 + m) * 2) = d2;
        *(v2f*)(base + (48 + m) * 2) = d3;
    }
}

// ---------------------------------------------------------------------------
// Kernel 2: s = X@a1, n = X@a2 for both components. One thread per row.
// Reads the pair-interleaved X layout.
// ---------------------------------------------------------------------------
__global__ void sn_kernel(const float* __restrict__ Xp_re, const float* __restrict__ Xp_im,
                          const float* __restrict__ a1,  const float* __restrict__ a2,
                          float* __restrict__ s_re, float* __restrict__ s_im,
                          float* __restrict__ n_re, float* __restrict__ n_im) {
    const int r = blockIdx.x * blockDim.x + threadIdx.x;
    if (r >= BN_) return;
    const float* xr = Xp_re + (long)(r >> 1) * (2 * U_) + (r & 1);
    const float* xi = Xp_im + (long)(r >> 1) * (2 * U_) + (r & 1);
    float sre = 0.f, sim = 0.f, nre = 0.f, nim = 0.f;
#pragma unroll 8
    for (int u = 0; u < U_; ++u) {
        const float vr = xr[u * 2], vi = xi[u * 2];
        const float c1 = a1[u], c2 = a2[u];
        sre = fmaf(vr, c1, sre); sim = fmaf(vi, c1, sim);
        nre = fmaf(vr, c2, nre); nim = fmaf(vi, c2, nim);
    }
    s_re[r] = sre; s_im[r] = sim; n_re[r] = nre; n_im[r] = nim;
}

// ---------------------------------------------------------------------------
// Kernel 3: per-row softmax statistics (max and 1/sum-exp), masked re + free im.
// One wave per row, b128-vectorized stream of the A row, online max/sum,
// butterfly shuffle reduction. This is the HBM-bound pass (A = 134 MB).
// ---------------------------------------------------------------------------
__global__ void stats_kernel(const float* __restrict__ A,
                             const float* __restrict__ s_re, const float* __restrict__ s_im,
                             const float* __restrict__ n_re, const float* __restrict__ n_im,
                             float* __restrict__ m1o, float* __restrict__ rz1o,
                             float* __restrict__ m2o, float* __restrict__ rz2o) {
    const int lane = threadIdx.x & 31;
    const int wave = threadIdx.x >> 5;
    const int r = blockIdx.x * (blockDim.x >> 5) + wave;    // row in [0, B*N)
    if (r >= BN_) return;
    const int b = r / N_;

    const float* __restrict__ arow = A    + (long)r * N_;
    const float* __restrict__ nre  = n_re + (long)b * N_;
    const float* __restrict__ nim  = n_im + (long)b * N_;
    const float sre = s_re[r], sim = s_im[r];

    float m1 = -INFINITY, z1 = 0.f;
    float m2 = -INFINITY, z2 = 0.f;

    for (int j0 = lane * 4; j0 < N_; j0 += 32 * 4) {
        v4f a  = *(const v4f*)(arow + j0);
        v4f v1 = *(const v4f*)(nre  + j0);
        v4f v2 = *(const v4f*)(nim  + j0);
#pragma unroll
        for (int t = 0; t < 4; ++t) {
            const float e1 = lrelu(sre + v1[t]) + NEG_BIG * (1.0f - a[t]);
            const float e2 = lrelu(sim + v2[t]);
            if (e1 > m1) { z1 = z1 * __expf(m1 - e1) + 1.0f; m1 = e1; }
            else         { z1 += __expf(e1 - m1); }
            if (e2 > m2) { z2 = z2 * __expf(m2 - e2) + 1.0f; m2 = e2; }
            else         { z2 += __expf(e2 - m2); }
        }
    }

    // wave32 butterfly reduction of (max, sum) pairs
#pragma unroll
    for (int off = 16; off >= 1; off >>= 1) {
        float om = __shfl_xor(m1, off, 32);
        float oz = __shfl_xor(z1, off, 32);
        float nm = fmaxf(m1, om);
        z1 = z1 * __expf(m1 - nm) + oz * __expf(om - nm);
        m1 = nm;
        om = __shfl_xor(m2, off, 32);
        oz = __shfl_xor(z2, off, 32);
        nm = fmaxf(m2, om);
        z2 = z2 * __expf(m2 - nm) + oz * __expf(om - nm);
        m2 = nm;
    }
    if (lane == 0) {
        m1o[r] = m1; rz1o[r] = 1.0f / z1;
        m2o[r] = m2; rz2o[r] = 1.0f / z2;
    }
}

// ---------------------------------------------------------------------------
// Kernel 4: fused alpha construction + complex matmul.
//   out_re = alpha1 @ Xre - alpha2 @ Xim
//   out_im = alpha1 @ Xim + alpha2 @ Xre
// One wave per 16-row i-tile. The A mask tile (16 x JCHUNK) is double-buffered
// into LDS with GLOBAL_LOAD_ASYNC_TO_LDS_B128 (ASYNCcnt pipe), so the hot loop's
// LOADcnt only covers the small Xp B-fragments. P1/P2 built in registers from
// the rank-1 logits + mask, then 16 f32 WMMAs per 4-j step.
// f32 WMMA has no A-neg modifier (NEG=CNeg only), so -P2 is formed in VALU.
// LDS: 32 KB n-vectors + 2 x 4 KB A ping-pong = 40 KB of the 320 KB WGP pool.
// ---------------------------------------------------------------------------
__global__ void __launch_bounds__(32)
attn_kernel(const float* __restrict__ A,
            const float* __restrict__ Xp_re, const float* __restrict__ Xp_im,
            const float* __restrict__ s_re, const float* __restrict__ s_im,
            const float* __restrict__ n_re, const float* __restrict__ n_im,
            const float* __restrict__ m1v, const float* __restrict__ rz1v,
            const float* __restrict__ m2v, const float* __restrict__ rz2v,
            float* __restrict__ out_re, float* __restrict__ out_im) {
    __shared__ float lnre[N_];
    __shared__ float lnim[N_];
    __shared__ float abuf[2][16 * JCHUNK];       // A tile ping-pong

    const int lane = threadIdx.x & 31;
    const int tile = blockIdx.x;                 // 0 .. B*N/16-1
    const int b    = tile / (N_ / 16);
    const int i0   = (tile % (N_ / 16)) * 16;
    const int m    = lane & 15;
    const int koff = (lane >> 4) * 2;

    const long r   = (long)b * N_ + i0 + m;      // this lane's row
    const float sre = s_re[r], sim = s_im[r];
    const float m1 = m1v[r], rz1 = rz1v[r];
    const float m2 = m2v[r], rz2 = rz2v[r];

    const float* __restrict__ atile = A     + ((long)b * N_ + i0) * N_;  // 16 rows
    const float* __restrict__ nre   = n_re  + (long)b * N_;
    const float* __restrict__ nim   = n_im  + (long)b * N_;
    const float* __restrict__ xre   = Xp_re + (long)b * N_ * U_;
    const float* __restrict__ xim   = Xp_im + (long)b * N_ * U_;

    // Stage the per-batch attention coefficient vectors in LDS (b128 copies).
    for (int t = lane * 4; t < N_; t += 32 * 4) {
        *(v4f*)(lnre + t) = *(const v4f*)(nre + t);
        *(v4f*)(lnim + t) = *(const v4f*)(nim + t);
    }
    __syncthreads();

    // Wave-relative LDS byte offsets for the async-DMA destinations
    // (flat LDS pointers truncate to the LDS aperture offset, addr[31:0]).
    const uint32_t abase = (uint32_t)(uintptr_t)(&abuf[0][0]);

    // Per-lane async segment assignment: chunk = 16 rows x JCHUNK floats
    // = 256 x 16B segments; segment s -> row = s>>4, 16B column (s&15).
    // Lane covers segments s = i*32 + lane, i = 0..SEGS-1.
    auto issue_chunk = [&](int jc, int buf) {
#pragma unroll
        for (int i = 0; i < SEGS; ++i) {
            const int s   = i * 32 + lane;
            const int row = s >> 4;
            const int c16 = s & 15;
            const uint64_t g = (uint64_t)(uintptr_t)(atile + (long)row * N_ + jc + c16 * 4);
            const uint32_t l = abase + (uint32_t)buf * (16 * JCHUNK * 4) + (uint32_t)s * 16;
            asm volatile("global_load_async_to_lds_b128 %0, %1, off"
                         :: "v"(l), "v"(g) : "memory");
        }
    };

    v8f accRe[4] = {v8f{}, v8f{}, v8f{}, v8f{}};
    v8f accIm[4] = {v8f{}, v8f{}, v8f{}, v8f{}};

    issue_chunk(0, 0);                            // prologue

    for (int c = 0; c < NCHUNK; ++c) {
        if (c + 1 < NCHUNK) {
            issue_chunk((c + 1) * JCHUNK, (c + 1) & 1);
            // in-order completion: oldest SEGS ops (= chunk c) have landed
            asm volatile("s_wait_asynccnt 8" ::: "memory");
        } else {
            asm volatile("s_wait_asynccnt 0" ::: "memory");
        }

        const float* __restrict__ ab = &abuf[c & 1][0];   // [row][JCHUNK]
        const int jc = c * JCHUNK;

#pragma unroll 2
        for (int jl = 0; jl < JCHUNK; jl += 4) {
            const int j = jc + jl + koff;

            v2f av = *(const v2f*)(ab + (m << 6) + jl + koff);  // ds_load_b64
            v2f nr = *(const v2f*)(lnre + j);                   // ds_load_b64
            v2f ni = *(const v2f*)(lnim + j);

            v2f p1, p2, p2n;
#pragma unroll
            for (int t = 0; t < 2; ++t) {
                const float e1 = lrelu(sre + nr[t]) + NEG_BIG * (1.0f - av[t]);
                const float e2 = lrelu(sim + ni[t]);
                const float a1 = __expf(e1 - m1) * rz1;   // alpha1[i, j+t]
                const float a2 = __expf(e2 - m2) * rz2;   // alpha2[i, j+t]
                p1[t]  = a1;
                p2[t]  = a2;
                p2n[t] = -a2;
            }

            // B fragments: one b64 per u-tile from the pair-interleaved layout.
            const long pbase = (long)(j >> 1) * (2 * U_) + m * 2;
            const float* xr0 = xre + pbase;
            const float* xi0 = xim + pbase;

#pragma unroll
            for (int ut = 0; ut < 4; ++ut) {
                const int uo = ut * 32;                   // 16 cols * 2 interleave
                v2f br = *(const v2f*)(xr0 + uo);         // {Xre[j][u], Xre[j+1][u]}
                v2f bi = *(const v2f*)(xi0 + uo);
                accRe[ut] = __builtin_amdgcn_wmma_f32_16x16x4_f32(false, p1,  false, br, (short)0, accRe[ut], false, false);
                accRe[ut] = __builtin_amdgcn_wmma_f32_16x16x4_f32(false, p2n, false, bi, (short)0, accRe[ut], false, false);
                accIm[ut] = __builtin_amdgcn_wmma_f32_16x16x4_f32(false, p1,  false, bi, (short)0, accIm[ut], false, false);
                accIm[ut] = __builtin_amdgcn_wmma_f32_16x16x4_f32(false, p2,  false, br, (short)0, accIm[ut], false, false);
            }
        }
    }

    const int rhi = (lane >> 4) * 8;
#pragma unroll
    for (int rr = 0; rr < 8; ++rr) {
        const long row = ((long)b * N_ + i0 + rhi + rr) * U_;
#pragma unroll
        for (int ut = 0; ut < 4; ++ut) {
            out_re[row + ut * 16 + m] = accRe[ut][rr];
            out_im[row + ut * 16 + m] = accIm[ut][rr];
        }
    }
}

// ---------------------------------------------------------------------------
// Host-side launcher
// Inputs (setup_inputs order): H_re, H_im, A, W, a_1, a_2  (all fp32)
// Output: out_re ++ out_im  (each [B,N,U] fp32)
// ---------------------------------------------------------------------------
extern "C" void kernel_launch(void* const* d_in, const int* in_sizes, int n_in,
                              void* d_out, int out_size, void* d_ws, size_t ws_size,
                              hipStream_t stream) {
    const float* Hre = (const float*)d_in[0];
    const float* Him = (const float*)d_in[1];
    const float* A   = (const float*)d_in[2];
    const float* W   = (const float*)d_in[3];
    const float* a1  = (const float*)d_in[4];
    const float* a2  = (const float*)d_in[5];

    float* out_re = (float*)d_out;
    float* out_im = out_re + (long)BN_ * U_;

    // Workspace layout (floats)
    float* ws    = (float*)d_ws;
    float* Xp_re = ws;                       // BN*U (pair-interleaved)
    float* Xp_im = Xp_re + (long)BN_ * U_;   // BN*U
    float* s_re  = Xp_im + (long)BN_ * U_;   // BN
    float* s_im  = s_re + BN_;
    float* n_re  = s_im + BN_;
    float* n_im  = n_re + BN_;
    float* m1    = n_im + BN_;
    float* rz1   = m1 + BN_;
    float* m2    = rz1 + BN_;
    float* rz2   = m2 + BN_;

    // 1) X = H @ W (WMMA), both components, pair-interleaved output
    xform_kernel<<<dim3(BN_ / 16, 2), 32, 0, stream>>>(Hre, Him, W, Xp_re, Xp_im);

    // 2) s / n projections
    sn_kernel<<<(BN_ + 255) / 256, 256, 0, stream>>>(Xp_re, Xp_im, a1, a2,
                                                     s_re, s_im, n_re, n_im);

    // 3) softmax row statistics (one wave per row; 8 waves per block)
    stats_kernel<<<BN_ / 8, 256, 0, stream>>>(A, s_re, s_im, n_re, n_im,
                                              m1, rz1, m2, rz2);

    // 4) fused alpha + complex matmul (WMMA + async-LDS A streaming)
    attn_kernel<<<BN_ / 16, 32, 0, stream>>>(A, Xp_re, Xp_im, s_re, s_im, n_re, n_im,
                                             m1, rz1, m2, rz2, out_re, out_im);
}